// spatialAttention_16518444221184
// MI455X (gfx1250) — compile-verified
//
#include <hip/hip_runtime.h>
#include <math.h>

typedef __attribute__((ext_vector_type(2))) float v2f;
typedef __attribute__((ext_vector_type(8))) float v8f;

#define TSTEPS 12
#define BATCH 16
#define NPTS 325
#define DMODEL 64
#define HEADS 8
#define DHEAD 8
#define ROWS1 (BATCH * NPTS)          // 5200
#define MT1 (ROWS1 / 16)              // 325 row tiles
#define ROWS2 (BATCH * TSTEPS * NPTS) // 62400
#define MT2 (ROWS2 / 16)              // 3900 row tiles
#define CT 21                         // ceil(325/16) key tiles
#define EPSV 1e-5f

static __device__ __forceinline__ v8f wmma4(v2f a, v2f b, v8f c) {
  // D = A(16x4,f32) * B(4x16,f32) + C(16x16,f32)
  return __builtin_amdgcn_wmma_f32_16x16x4_f32(false, a, false, b, (short)0, c,
                                               false, false);
}

// ---------------------------------------------------------------------------
// Kernel 1: per-t QKV projection  Y[t, m, :] = Xc[t, m, :] @ W_qkv + b_qkv
//   Xc = concat(X, STE) along features (128). One wave per 16-row tile.
// ---------------------------------------------------------------------------
__global__ __launch_bounds__(256) void k_qkv_gemm(
    const float* __restrict__ X, const float* __restrict__ STE,
    const float* __restrict__ Wqkv, const float* __restrict__ bqkv,
    float* __restrict__ Y) {
  const int t = blockIdx.y;
  const int wave = threadIdx.x >> 5;
  const int lane = threadIdx.x & 31;
  const int mtile = blockIdx.x * 8 + wave;
  if (mtile >= MT1) return;
  const int hi = lane >> 4, ln = lane & 15;

  const int m = mtile * 16 + ln;  // A-frag row (flattened b*N+n)
  const int bb = m / NPTS, nn = m % NPTS;
  const long xbase = ((long)(bb * TSTEPS + t) * NPTS + nn) * DMODEL;

  v8f acc[4] = {};
  for (int kk = 0; kk < 32; ++kk) {
    const int c = kk * 4 + 2 * hi;  // k index this half-lane covers (c, c+1)
    v2f a;
    if (c < 64) {
      a.x = X[xbase + c];
      a.y = X[xbase + c + 1];
    } else {
      a.x = STE[xbase + c - 64];
      a.y = STE[xbase + c - 63];
    }
#pragma unroll
    for (int nt = 0; nt < 4; ++nt) {
      v2f wb;
      wb.x = Wqkv[c * DMODEL + nt * 16 + ln];
      wb.y = Wqkv[(c + 1) * DMODEL + nt * 16 + ln];
      acc[nt] = wmma4(a, wb, acc[nt]);
    }
  }
#pragma unroll
  for (int nt = 0; nt < 4; ++nt)
#pragma unroll
    for (int g = 0; g < 8; ++g) {
      const int row = g + 8 * hi;
      const int col = nt * 16 + ln;
      Y[((long)t * ROWS1 + mtile * 16 + row) * DMODEL + col] =
          acc[nt][g] + bqkv[col];
    }
}

// ---------------------------------------------------------------------------
// Kernel 2: BN1 stats over (B,N) per (t,d) -> scale/shift
// ---------------------------------------------------------------------------
__global__ void k_bn1_stats(const float* __restrict__ Y,
                            const float* __restrict__ g,
                            const float* __restrict__ be,
                            float* __restrict__ sc, float* __restrict__ sh) {
  const int d = blockIdx.x, t = blockIdx.y, tid = threadIdx.x;
  __shared__ float ss[256], sq[256];
  float s = 0.f, q = 0.f;
  for (int i = tid; i < ROWS1; i += 256) {
    const float v = Y[((long)t * ROWS1 + i) * DMODEL + d];
    s += v;
    q += v * v;
  }
  ss[tid] = s;
  sq[tid] = q;
  __syncthreads();
  for (int o = 128; o > 0; o >>= 1) {
    if (tid < o) {
      ss[tid] += ss[tid + o];
      sq[tid] += sq[tid + o];
    }
    __syncthreads();
  }
  if (tid == 0) {
    const float mean = ss[0] / (float)ROWS1;
    const float var = sq[0] / (float)ROWS1 - mean * mean;
    const float scale = g[d] / sqrtf(var + EPSV);
    sc[t * DMODEL + d] = scale;
    sh[t * DMODEL + d] = be[d] - mean * scale;
  }
}

// ---------------------------------------------------------------------------
// Kernel 3: BN1 apply + ReLU (in place) -> qkv tensor
// ---------------------------------------------------------------------------
__global__ void k_bn1_apply(float* __restrict__ Y, const float* __restrict__ sc,
                            const float* __restrict__ sh) {
  const long idx = (long)blockIdx.x * 256 + threadIdx.x;
  if (idx >= (long)TSTEPS * ROWS1 * DMODEL) return;
  const int d = (int)(idx & 63);
  const int t = (int)(idx / ((long)ROWS1 * DMODEL));
  const float v = Y[idx] * sc[t * DMODEL + d] + sh[t * DMODEL + d];
  Y[idx] = fmaxf(v, 0.f);
}

// ---------------------------------------------------------------------------
// Kernel 4: fused attention + per-step linear.
//   Grid (rowtile, b, step), block = 8 waves (= 8 heads).
//   Each wave: online-softmax attention over j=0..step, immediately projects
//   each head's O(16x8) through lin_w[step, j*64+h*8 : +8, :] and accumulates.
//   Heads reduced via LDS, Xo[b, step, n, :] written directly.
// ---------------------------------------------------------------------------
__global__ __launch_bounds__(256) void k_attn(const float* __restrict__ QKV,
                                              const float* __restrict__ linw,
                                              const float* __restrict__ linb,
                                              float* __restrict__ Xo) {
  const int rt = blockIdx.x;    // query row tile
  const int b = blockIdx.y;
  const int step = blockIdx.z;
  const int wave = threadIdx.x >> 5;  // = head
  const int lane = threadIdx.x & 31;
  const int hi = lane >> 4, ln = lane & 15;
  const int h = wave;
  const float scale = 0.35355339059327373f;  // 1/sqrt(8)
  const float NEG_INF = -__builtin_inff();

  __shared__ float trbuf[8][16][17];  // per-wave C-frag -> A-frag transpose
  __shared__ float red[16][64];       // cross-head reduction
  for (int i = threadIdx.x; i < 16 * 64; i += 256) red[i >> 6][i & 63] = 0.f;
  __syncthreads();

  // Q A-fragments for k = 0..7 (d_head)
  const int qrow = min(rt * 16 + ln, NPTS - 1);
  const long qoff = ((long)step * ROWS1 + b * NPTS + qrow) * DMODEL + h * DHEAD;
  v2f qa0, qa1;
  qa0.x = QKV[qoff + 2 * hi];
  qa0.y = QKV[qoff + 2 * hi + 1];
  qa1.x = QKV[qoff + 4 + 2 * hi];
  qa1.y = QKV[qoff + 4 + 2 * hi + 1];

  v8f acc[4] = {};  // accumulated 16x64 projected output

  for (int j = 0; j <= step; ++j) {
    const long kvbase = ((long)j * ROWS1 + b * NPTS) * DMODEL + h * DHEAD;
    float mrow[8], lrow[8], p[8];
    v8f oacc = {};
#pragma unroll
    for (int g = 0; g < 8; ++g) {
      mrow[g] = NEG_INF;
      lrow[g] = 0.f;
    }

    for (int ct = 0; ct < CT; ++ct) {
      // --- S tile = Q @ K^T (scaled, masked) ---
      const int key = ct * 16 + ln;
      const int keyc = min(key, NPTS - 1);
      const long ko = kvbase + (long)keyc * DMODEL;
      v2f kb0, kb1;
      kb0.x = QKV[ko + 2 * hi];
      kb0.y = QKV[ko + 2 * hi + 1];
      kb1.x = QKV[ko + 4 + 2 * hi];
      kb1.y = QKV[ko + 4 + 2 * hi + 1];
      v8f s = {};
      s = wmma4(qa0, kb0, s);
      s = wmma4(qa1, kb1, s);
      const bool valid = key < NPTS;

      // --- online softmax update (per row = per C-frag VGPR, per half) ---
#pragma unroll
      for (int g = 0; g < 8; ++g) {
        const float sv = valid ? s[g] * scale : NEG_INF;
        float t2 = sv;
#pragma unroll
        for (int off = 1; off <= 8; off <<= 1)
          t2 = fmaxf(t2, __shfl_xor(t2, off, 32));
        const float newm = fmaxf(mrow[g], t2);
        const float alpha = __expf(mrow[g] - newm);
        const float pv = __expf(sv - newm);
        float sum = pv;
#pragma unroll
        for (int off = 1; off <= 8; off <<= 1) sum += __shfl_xor(sum, off, 32);
        lrow[g] = lrow[g] * alpha + sum;
        mrow[g] = newm;
        oacc[g] *= alpha;
        p[g] = pv;
      }

      // --- O += P @ V : transpose P (C layout -> A layout) via LDS ---
#pragma unroll
      for (int g = 0; g < 8; ++g) trbuf[wave][g + 8 * hi][ln] = p[g];
#pragma unroll
      for (int kk = 0; kk < 4; ++kk) {
        const int kidx = 4 * kk + 2 * hi;
        v2f pa, vb;
        pa.x = trbuf[wave][ln][kidx];
        pa.y = trbuf[wave][ln][kidx + 1];
        const int mm = min(ct * 16 + kidx, NPTS - 1);
        const int mm1 = min(ct * 16 + kidx + 1, NPTS - 1);
        const int dcol = ln & 7;  // only d<8 columns are meaningful
        vb.x = QKV[kvbase + (long)mm * DMODEL + dcol];
        vb.y = QKV[kvbase + (long)mm1 * DMODEL + dcol];
        oacc = wmma4(pa, vb, oacc);
      }
    }  // key tiles

    // normalize O by softmax denominator
#pragma unroll
    for (int g = 0; g < 8; ++g) oacc[g] /= lrow[g];

    // --- project: acc += O(16x8) @ lin_w[step, j*64+h*8 : +8, :] ---
#pragma unroll
    for (int g = 0; g < 8; ++g) trbuf[wave][g + 8 * hi][ln] = oacc[g];
    const long wbase = ((long)step * (TSTEPS * DMODEL) + j * DMODEL + h * DHEAD) *
                       DMODEL;
#pragma unroll
    for (int kk = 0; kk < 2; ++kk) {
      const int kidx = 4 * kk + 2 * hi;  // k in [0,8)
      v2f oa;
      oa.x = trbuf[wave][ln][kidx];
      oa.y = trbuf[wave][ln][kidx + 1];
#pragma unroll
      for (int ot = 0; ot < 4; ++ot) {
        v2f wb;
        wb.x = linw[wbase + (long)kidx * DMODEL + ot * 16 + ln];
        wb.y = linw[wbase + (long)(kidx + 1) * DMODEL + ot * 16 + ln];
        acc[ot] = wmma4(oa, wb, acc[ot]);
      }
    }
  }  // j

  // --- reduce 8 heads, add bias, store Xo[b, step, n, :] ---
  __syncthreads();
#pragma unroll
  for (int ot = 0; ot < 4; ++ot)
#pragma unroll
    for (int g = 0; g < 8; ++g)
      atomicAdd(&red[g + 8 * hi][ot * 16 + ln], acc[ot][g]);
  __syncthreads();
  for (int i = threadIdx.x; i < 16 * 64; i += 256) {
    const int row = i >> 6, col = i & 63;
    const int grow = rt * 16 + row;
    if (grow < NPTS)
      Xo[((long)(b * TSTEPS + step) * NPTS + grow) * DMODEL + col] =
          red[row][col] + linb[step * DMODEL + col];
  }
}

// ---------------------------------------------------------------------------
// Kernel 5: final FC  Ypre = Xo @ W_fc + b_fc  (62400 x 64 @ 64 x 64)
// ---------------------------------------------------------------------------
__global__ __launch_bounds__(256) void k_fc_gemm(const float* __restrict__ Xo,
                                                 const float* __restrict__ Wfc,
                                                 const float* __restrict__ bfc,
                                                 float* __restrict__ Y) {
  const int wave = threadIdx.x >> 5, lane = threadIdx.x & 31;
  const int mtile = blockIdx.x * 8 + wave;
  if (mtile >= MT2) return;
  const int hi = lane >> 4, ln = lane & 15;
  const long m = (long)mtile * 16 + ln;

  v8f acc[4] = {};
  for (int kk = 0; kk < 16; ++kk) {
    const int c = kk * 4 + 2 * hi;
    v2f a;
    a.x = Xo[m * DMODEL + c];
    a.y = Xo[m * DMODEL + c + 1];
#pragma unroll
    for (int nt = 0; nt < 4; ++nt) {
      v2f wb;
      wb.x = Wfc[c * DMODEL + nt * 16 + ln];
      wb.y = Wfc[(c + 1) * DMODEL + nt * 16 + ln];
      acc[nt] = wmma4(a, wb, acc[nt]);
    }
  }
#pragma unroll
  for (int nt = 0; nt < 4; ++nt)
#pragma unroll
    for (int g = 0; g < 8; ++g) {
      const long row = (long)mtile * 16 + g + 8 * hi;
      const int col = nt * 16 + ln;
      Y[row * DMODEL + col] = acc[nt][g] + bfc[col];
    }
}

// ---------------------------------------------------------------------------
// Kernel 6: BN2 stats over (B,T,N) per d
// ---------------------------------------------------------------------------
__global__ void k_bn2_stats(const float* __restrict__ Y,
                            const float* __restrict__ g,
                            const float* __restrict__ be,
                            float* __restrict__ sc, float* __restrict__ sh) {
  const int d = blockIdx.x, tid = threadIdx.x;
  __shared__ float ss[256], sq[256];
  float s = 0.f, q = 0.f;
  for (int i = tid; i < ROWS2; i += 256) {
    const float v = Y[(long)i * DMODEL + d];
    s += v;
    q += v * v;
  }
  ss[tid] = s;
  sq[tid] = q;
  __syncthreads();
  for (int o = 128; o > 0; o >>= 1) {
    if (tid < o) {
      ss[tid] += ss[tid + o];
      sq[tid] += sq[tid + o];
    }
    __syncthreads();
  }
  if (tid == 0) {
    const float mean = ss[0] / (float)ROWS2;
    const float var = sq[0] / (float)ROWS2 - mean * mean;
    const float scale = g[d] / sqrtf(var + EPSV);
    sc[d] = scale;
    sh[d] = be[d] - mean * scale;
  }
}

// ---------------------------------------------------------------------------
// Kernel 7: BN2 apply + ReLU -> d_out
// ---------------------------------------------------------------------------
__global__ void k_bn2_apply(const float* __restrict__ Y,
                            const float* __restrict__ sc,
                            const float* __restrict__ sh,
                            float* __restrict__ out) {
  const long idx = (long)blockIdx.x * 256 + threadIdx.x;
  if (idx >= (long)ROWS2 * DMODEL) return;
  const int d = (int)(idx & 63);
  out[idx] = fmaxf(Y[idx] * sc[d] + sh[d], 0.f);
}

// ---------------------------------------------------------------------------
extern "C" void kernel_launch(void* const* d_in, const int* in_sizes, int n_in,
                              void* d_out, int out_size, void* d_ws,
                              size_t ws_size, hipStream_t stream) {
  (void)in_sizes; (void)n_in; (void)out_size; (void)ws_size;
  const float* X = (const float*)d_in[0];
  const float* STE = (const float*)d_in[1];
  const float* Wqkv = (const float*)d_in[2];
  const float* bqkv = (const float*)d_in[3];
  const float* gqkv = (const float*)d_in[4];
  const float* beqkv = (const float*)d_in[5];
  const float* linw = (const float*)d_in[6];
  const float* linb = (const float*)d_in[7];
  const float* Wfc = (const float*)d_in[8];
  const float* bfc = (const float*)d_in[9];
  const float* gfc = (const float*)d_in[10];
  const float* befc = (const float*)d_in[11];
  float* out = (float*)d_out;

  // workspace layout (bytes): qkv 15.97MB | Xo 15.97MB | Ypre 15.97MB | stats
  const size_t SEG = (size_t)TSTEPS * ROWS1 * DMODEL * sizeof(float);  // 15974400
  char* ws = (char*)d_ws;
  float* Yqkv = (float*)(ws);
  float* Xo = (float*)(ws + SEG);
  float* Ypre = (float*)(ws + 2 * SEG);
  float* sc1 = (float*)(ws + 3 * SEG);
  float* sh1 = sc1 + TSTEPS * DMODEL;
  float* sc2 = sh1 + TSTEPS * DMODEL;
  float* sh2 = sc2 + DMODEL;

  hipLaunchKernelGGL(k_qkv_gemm, dim3((MT1 + 7) / 8, TSTEPS), dim3(256), 0,
                     stream, X, STE, Wqkv, bqkv, Yqkv);
  hipLaunchKernelGGL(k_bn1_stats, dim3(DMODEL, TSTEPS), dim3(256), 0, stream,
                     Yqkv, gqkv, beqkv, sc1, sh1);
  hipLaunchKernelGGL(k_bn1_apply,
                     dim3((unsigned)(((long)TSTEPS * ROWS1 * DMODEL + 255) / 256)),
                     dim3(256), 0, stream, Yqkv, sc1, sh1);
  hipLaunchKernelGGL(k_attn, dim3(CT, BATCH, TSTEPS), dim3(256), 0, stream,
                     Yqkv, linw, linb, Xo);
  hipLaunchKernelGGL(k_fc_gemm, dim3((MT2 + 7) / 8), dim3(256), 0, stream, Xo,
                     Wfc, bfc, Ypre);
  hipLaunchKernelGGL(k_bn2_stats, dim3(DMODEL), dim3(256), 0, stream, Ypre, gfc,
                     befc, sc2, sh2);
  hipLaunchKernelGGL(k_bn2_apply,
                     dim3((unsigned)(((long)ROWS2 * DMODEL + 255) / 256)),
                     dim3(256), 0, stream, Ypre, sc2, sh2, out);
}